// GCN1_27032524161268
// MI455X (gfx1250) — compile-verified
//
#include <hip/hip_runtime.h>
#include <hip/hip_bf16.h>
#include <math.h>

typedef __attribute__((ext_vector_type(2))) float v2f;
typedef __attribute__((ext_vector_type(8))) float v8f;

#define N_NODES 100000
#define N_FEAT  512
#define N_CLASS 40
#define N_EDGES 3200000
#define NPAD    48      // 40 padded to 3 WMMA tiles of 16
#define CHUNK_K 256     // K chunk of W staged in LDS (256*48*4 = 48 KB)

// ---------------------------------------------------------------- init
__global__ void k_init(float* __restrict__ deg, float* __restrict__ acc) {
  int i = blockIdx.x * blockDim.x + threadIdx.x;
  if (i < N_NODES) deg[i] = 1.0f;   // self-loop contributes 1 to degree
  long total = (long)N_NODES * N_CLASS;
  long stride = (long)gridDim.x * blockDim.x;
  for (long j = i; j < total; j += stride) acc[j] = 0.0f;
}

// ---------------------------------------------------------------- GEMM h = x @ W
// block = 256 threads = 8 waves; each wave -> 16 rows x 48 cols (3 WMMA N-tiles)
// LDS holds W K-pair-interleaved: lds[(p*NPAD + n)*2 + {0,1}] = W[2p][n], W[2p+1][n]
// so each 4x16 B fragment is ONE contiguous ds_load_b64 per lane (no repack movs).
__global__ __launch_bounds__(256)
void k_gemm(const float* __restrict__ x, const float* __restrict__ W,
            float* __restrict__ h) {
  __shared__ float lds_w[CHUNK_K * NPAD];   // 48 KB (CHUNK_K/2 pair-rows * NPAD * 2)

  const int wave = threadIdx.x >> 5;
  const int lane = threadIdx.x & 31;
  const int g    = lane >> 4;       // half-wave (selects K pair)
  const int lm   = lane & 15;

  const int rowBase = blockIdx.x * 128 + wave * 16;
  int row  = rowBase + lm;
  int rowC = row < N_NODES ? row : N_NODES - 1;   // clamp: keep EXEC full
  const float* xrow = x + (long)rowC * N_FEAT;

  v8f acc0 = {}, acc1 = {}, acc2 = {};

  for (int kc = 0; kc < N_FEAT; kc += CHUNK_K) {
    __syncthreads();
    // stage W[kc..kc+CHUNK_K) x 48 (zero-padded cols 40..47), pair-interleaved
    for (int idx = threadIdx.x; idx < (CHUNK_K / 2) * NPAD; idx += 256) {
      int p  = idx / NPAD;          // K pair row
      int nn = idx % NPAD;
      int k0 = kc + 2 * p;
      v2f w;
      w.x = (nn < N_CLASS) ? W[(k0 + 0) * N_CLASS + nn] : 0.0f;
      w.y = (nn < N_CLASS) ? W[(k0 + 1) * N_CLASS + nn] : 0.0f;
      *(v2f*)&lds_w[idx * 2] = w;   // ds_store_b64
    }
    __syncthreads();

    for (int k = 0; k < CHUNK_K; k += 4) {
      // A fragment: 16x4 f32; lane half g holds K = k+2g, k+2g+1 for row lm
      v2f a = *(const v2f*)(xrow + kc + k + 2 * g);
      // B fragments: pair-row p = k/2 + g holds W[k+2g], W[k+2g+1] contiguously
      const float* bp = &lds_w[((k / 2 + g) * NPAD + lm) * 2];
      v2f b0 = *(const v2f*)(bp +  0);   // cols  0..15
      v2f b1 = *(const v2f*)(bp + 32);   // cols 16..31
      v2f b2 = *(const v2f*)(bp + 64);   // cols 32..47
      acc0 = __builtin_amdgcn_wmma_f32_16x16x4_f32(false, a, false, b0, (short)0, acc0, false, false);
      acc1 = __builtin_amdgcn_wmma_f32_16x16x4_f32(false, a, false, b1, (short)0, acc1, false, false);
      acc2 = __builtin_amdgcn_wmma_f32_16x16x4_f32(false, a, false, b2, (short)0, acc2, false, false);
    }
  }

  // C/D layout: VGPR v -> row rowBase + v + 8*g, col = tile*16 + lm
  #pragma unroll
  for (int v = 0; v < 8; ++v) {
    int r = rowBase + v + 8 * g;
    if (r < N_NODES) {
      float* hr = h + (long)r * N_CLASS;
      hr[lm]      = acc0[v];
      hr[lm + 16] = acc1[v];
      if (lm + 32 < N_CLASS) hr[lm + 32] = acc2[v];
    }
  }
}

// ---------------------------------------------------------------- degree (atomic)
__global__ void k_degree(const int* __restrict__ dst, float* __restrict__ deg) {
  int e = blockIdx.x * blockDim.x + threadIdx.x;
  if (e < N_EDGES) atomicAdd(&deg[dst[e]], 1.0f);
}

// ---------------------------------------------------------------- dinv = rsqrt(deg)
__global__ void k_rsqrt(float* __restrict__ deg) {
  int i = blockIdx.x * blockDim.x + threadIdx.x;
  if (i < N_NODES) deg[i] = rsqrtf(deg[i]);   // deg >= 1 always (self-loop)
}

// ---------------------------------------------------------------- edge scatter
__global__ __launch_bounds__(256)
void k_scatter(const int* __restrict__ src, const int* __restrict__ dst,
               const float* __restrict__ dinv, const float* __restrict__ h,
               float* __restrict__ acc) {
  int e = blockIdx.x * blockDim.x + threadIdx.x;
  if (e >= N_EDGES) return;
  int s = src[e], d = dst[e];
  float norm = dinv[s] * dinv[d];
  const float4* hs = (const float4*)(h + (long)s * N_CLASS);  // 160B row, 16B aligned
  float* ad = acc + (long)d * N_CLASS;
  #pragma unroll
  for (int q = 0; q < N_CLASS / 4; ++q) {
    float4 m = hs[q];
    atomicAdd(ad + 4 * q + 0, m.x * norm);
    atomicAdd(ad + 4 * q + 1, m.y * norm);
    atomicAdd(ad + 4 * q + 2, m.z * norm);
    atomicAdd(ad + 4 * q + 3, m.w * norm);
  }
}

// ---------------------------------------------------------------- self-loop + bias + log_softmax
__global__ void k_finalize(const float* __restrict__ acc, const float* __restrict__ h,
                           const float* __restrict__ dinv, const float* __restrict__ bias,
                           float* __restrict__ out) {
  int i = blockIdx.x * blockDim.x + threadIdx.x;
  if (i >= N_NODES) return;
  float di = dinv[i];
  float self = di * di;
  const float* ai = acc + (long)i * N_CLASS;
  const float* hi = h + (long)i * N_CLASS;
  float v[N_CLASS];
  float m = -INFINITY;
  #pragma unroll
  for (int c = 0; c < N_CLASS; ++c) {
    float t = ai[c] + hi[c] * self + bias[c];
    v[c] = t;
    m = fmaxf(m, t);
  }
  float ssum = 0.0f;
  #pragma unroll
  for (int c = 0; c < N_CLASS; ++c) ssum += expf(v[c] - m);
  float lse = m + logf(ssum);
  float* oi = out + (long)i * N_CLASS;
  #pragma unroll
  for (int c = 0; c < N_CLASS; ++c) oi[c] = v[c] - lse;
}

// ---------------------------------------------------------------- launch
extern "C" void kernel_launch(void* const* d_in, const int* in_sizes, int n_in,
                              void* d_out, int out_size, void* d_ws, size_t ws_size,
                              hipStream_t stream) {
  (void)in_sizes; (void)n_in; (void)out_size; (void)ws_size;
  const float* x    = (const float*)d_in[0];
  const int*   edge = (const int*)d_in[1];     // [2, E] flat: row0 = src, row1 = dst
  const float* W    = (const float*)d_in[2];
  const float* b    = (const float*)d_in[3];
  const int* src = edge;
  const int* dst = edge + N_EDGES;
  float* out = (float*)d_out;

  // workspace: h [N*40] | deg/dinv [N] | acc [N*40]  (~32.4 MB)
  float* h   = (float*)d_ws;
  float* deg = h + (size_t)N_NODES * N_CLASS;
  float* acc = deg + N_NODES;

  k_init    <<<1024, 256, 0, stream>>>(deg, acc);
  k_gemm    <<<(N_NODES + 127) / 128, 256, 0, stream>>>(x, W, h);
  k_degree  <<<(N_EDGES + 255) / 256, 256, 0, stream>>>(dst, deg);
  k_rsqrt   <<<(N_NODES + 255) / 256, 256, 0, stream>>>(deg);
  k_scatter <<<(N_EDGES + 255) / 256, 256, 0, stream>>>(src, dst, deg, h, acc);
  k_finalize<<<(N_NODES + 255) / 256, 256, 0, stream>>>(acc, h, deg, b, out);
}